// LightGCN_2860448219519
// MI455X (gfx1250) — compile-verified
//
#include <hip/hip_runtime.h>
#include <hip/hip_bf16.h>

#ifndef __has_builtin
#define __has_builtin(x) 0
#endif

#define AS1 __attribute__((address_space(1)))
#define AS3 __attribute__((address_space(3)))

#define DDIM 64          // embedding dim (reference D)
#define TILE 256         // edges staged per block tile
#define SPMM_BLOCK 256   // 8 wave32 per block, 16 half-waves

// ---------------- CDNA5 async global->LDS staging (ASYNCcnt path) ----------
__device__ __forceinline__ void async_ld_b32(const void* g, void* l) {
#if __has_builtin(__builtin_amdgcn_global_load_async_to_lds_b32)
    __builtin_amdgcn_global_load_async_to_lds_b32((AS1 int*)g, (AS3 int*)l, 0, 0);
#else
    *(int*)l = *(const int*)g;   // synchronous fallback
#endif
}

template <int N>
__device__ __forceinline__ void wait_async() {
#if __has_builtin(__builtin_amdgcn_s_wait_asynccnt)
    __builtin_amdgcn_s_wait_asynccnt(N);     // constant immediate
#else
    asm volatile("s_wait_asynccnt %0" :: "i"(N) : "memory");
#endif
}

// ---------------- no-return f32 global atomic (rides STOREcnt) -------------
__device__ __forceinline__ void gatomic_add(float* p, float v) {
#if __has_builtin(__builtin_amdgcn_global_atomic_fadd_f32)
    (void)__builtin_amdgcn_global_atomic_fadd_f32((AS1 float*)p, v);
#else
    unsafeAtomicAdd(p, v);   // lowers to global_atomic_add_f32 (no CAS loop)
#endif
}

// ---------------------------------------------------------------------------
// Dual SpMM over the shared edge list:
//   dstU[r] += v * srcI[c]   (users aggregate item messages)
//   dstI[r] += v * srcU[c]   (items aggregate user messages)
// Half-wave (16 lanes) owns one edge; each lane moves one float4 (16B) of the
// 256B row -> fully coalesced b128 gathers + contiguous f32 atomic scatters.
// Edge records double-buffered in LDS via async loads (ASYNCcnt).
//
// IMPORTANT: staging uses a clamped index (not a per-lane branch) so every
// wave issues exactly 3 async-load instructions per stage; this keeps the
// per-wave ASYNCcnt bookkeeping (<=3 while prefetching, <=0 on the tail)
// exact even for the final partial tile. Tail entries load valid-but-unused
// data; the consume loop bound-checks e < nE.
// ---------------------------------------------------------------------------
__global__ void __launch_bounds__(SPMM_BLOCK)
spmm_dual(const float* __restrict__ srcU, const float* __restrict__ srcI,
          float* __restrict__ dstU, float* __restrict__ dstI,
          const float* __restrict__ eval, const int* __restrict__ erow,
          const int* __restrict__ ecol, int nE, int nTiles)
{
    __shared__ float s_val[2][TILE];
    __shared__ int   s_row[2][TILE];
    __shared__ int   s_col[2][TILE];

    const int t   = threadIdx.x;
    const int hw  = t >> 4;    // half-wave id within block (0..15)
    const int sub = t & 15;    // float4 slot within the 64-float row

    auto stage = [&](int tile, int b) {
        int e  = tile * TILE + t;
        int ec = e < nE ? e : (nE - 1);      // clamp: keep EXEC all-ones
        async_ld_b32(eval + ec, &s_val[b][t]);
        async_ld_b32(erow + ec, &s_row[b][t]);
        async_ld_b32(ecol + ec, &s_col[b][t]);
    };

    int tile = (int)blockIdx.x;
    if (tile < nTiles) stage(tile, 0);
    int buf = 0;

    while (tile < nTiles) {
        int next = tile + (int)gridDim.x;
        if (next < nTiles) { stage(next, buf ^ 1); wait_async<3>(); }
        else               { wait_async<0>(); }
        __syncthreads();

        const int base = tile * TILE;
        for (int k = 0; k < TILE / 16; ++k) {
            const int idx = (hw << 4) + k;
            const int e   = base + idx;
            if (e >= nE) break;
            const float v = s_val[buf][idx];   // LDS broadcast within half-wave
            const int   r = s_row[buf][idx];
            const int   c = s_col[buf][idx];

            const float4 xi = ((const float4*)(srcI + (size_t)c * DDIM))[sub];
            const float4 xu = ((const float4*)(srcU + (size_t)c * DDIM))[sub];

            float* du = dstU + (size_t)r * DDIM + sub * 4;
            float* di = dstI + (size_t)r * DDIM + sub * 4;
            gatomic_add(du + 0, v * xi.x); gatomic_add(du + 1, v * xi.y);
            gatomic_add(du + 2, v * xi.z); gatomic_add(du + 3, v * xi.w);
            gatomic_add(di + 0, v * xu.x); gatomic_add(di + 1, v * xu.y);
            gatomic_add(di + 2, v * xu.z); gatomic_add(di + 3, v * xu.w);
        }
        __syncthreads();
        buf ^= 1;
        tile = next;
    }
}

// ------------------------- epilogue / init kernels -------------------------
__global__ void init_zero(float4* __restrict__ out, float4* __restrict__ bufA, long n4)
{
    long i = (long)blockIdx.x * blockDim.x + threadIdx.x;
    const long s = (long)gridDim.x * blockDim.x;
    const float4 z = {0.f, 0.f, 0.f, 0.f};
    for (; i < n4; i += s) { out[i] = z; bufA[i] = z; }
}

__global__ void acc_zero(float4* __restrict__ out, const float4* __restrict__ src,
                         float4* __restrict__ zbuf, long n4)
{
    long i = (long)blockIdx.x * blockDim.x + threadIdx.x;
    const long s = (long)gridDim.x * blockDim.x;
    const float4 z = {0.f, 0.f, 0.f, 0.f};
    for (; i < n4; i += s) {
        float4 o = out[i]; float4 a = src[i];
        o.x += a.x; o.y += a.y; o.z += a.z; o.w += a.w;
        out[i] = o; zbuf[i] = z;
    }
}

__global__ void acc_final(float4* __restrict__ out, const float4* __restrict__ src, long n4)
{
    long i = (long)blockIdx.x * blockDim.x + threadIdx.x;
    const long s = (long)gridDim.x * blockDim.x;
    const float inv3 = 1.0f / 3.0f;
    for (; i < n4; i += s) {
        float4 o = out[i]; float4 a = src[i];
        o.x = (o.x + a.x) * inv3; o.y = (o.y + a.y) * inv3;
        o.z = (o.z + a.z) * inv3; o.w = (o.w + a.w) * inv3;
        out[i] = o;
    }
}

// ---------------------------------------------------------------------------
extern "C" void kernel_launch(void* const* d_in, const int* in_sizes, int n_in,
                              void* d_out, int out_size, void* d_ws, size_t ws_size,
                              hipStream_t stream)
{
    const float* user_emb = (const float*)d_in[0];   // N*D f32
    const float* item_emb = (const float*)d_in[1];   // N*D f32
    const float* eval     = (const float*)d_in[2];   // E f32
    const int*   erow     = (const int*)d_in[3];     // E i32
    const int*   ecol     = (const int*)d_in[4];     // E i32

    const long ND = (long)in_sizes[0];               // N*D = 12.8M
    const int  nE = in_sizes[2];                     // 6.4M

    float* out = (float*)d_out;                      // [u_final(ND) | i_final(ND)]
    float* ws  = (float*)d_ws;                       // needs 4*ND floats (~205 MB)
    float* uA = ws;            float* iA = ws + ND;
    float* uB = ws + 2 * ND;   float* iB = ws + 3 * ND;
    float* bufA = uA;          float* bufB = uB;     // each is a contiguous 2*ND block

    const long n2 = 2 * ND, n4 = n2 / 4;
    const int  TB = 256;
    long gU = (n4 + TB - 1) / TB; if (gU > 32768) gU = 32768;
    const int gridU = (int)gU;

    const int nTiles = (nE + TILE - 1) / TILE;
    const int gridS  = nTiles < 4096 ? nTiles : 4096; // persistent-ish, enables dbl-buffer

    if (nE <= 0 || ND <= 0) return;

    // layer 1: u1 = A@i0, i1 = A@u0
    init_zero<<<gridU, TB, 0, stream>>>((float4*)out, (float4*)bufA, n4);
    spmm_dual<<<gridS, SPMM_BLOCK, 0, stream>>>(user_emb, item_emb, uA, iA,
                                                eval, erow, ecol, nE, nTiles);
    acc_zero<<<gridU, TB, 0, stream>>>((float4*)out, (const float4*)bufA, (float4*)bufB, n4);

    // layer 2: u2 = A@i1, i2 = A@u1
    spmm_dual<<<gridS, SPMM_BLOCK, 0, stream>>>(uA, iA, uB, iB,
                                                eval, erow, ecol, nE, nTiles);
    acc_zero<<<gridU, TB, 0, stream>>>((float4*)out, (const float4*)bufB, (float4*)bufA, n4);

    // layer 3: u3 = A@i2, i3 = A@u2
    spmm_dual<<<gridS, SPMM_BLOCK, 0, stream>>>(uB, iB, uA, iA,
                                                eval, erow, ecol, nE, nTiles);
    acc_final<<<gridU, TB, 0, stream>>>((float4*)out, (const float4*)bufA, n4);
}